// DifferentiableRankIntegration_67954972557746
// MI455X (gfx1250) — compile-verified
//
#include <hip/hip_runtime.h>
#include <stdint.h>

// ---------------------------------------------------------------------------
// DifferentiableRankIntegration for MI455X (gfx1250, wave32).
//
// rank reduction rewritten as  sig(k,j) = e_k / (e_k + e_j),  e = exp(s/tau)
// (exact identity) so the 2^30-per-matrix inner loop needs only
// add + v_rcp_f32 + mul + add + fma per pair and exp is hoisted to O(B^2).
// Rows are staged into LDS with gfx1250 async-to-LDS ops (ASYNCcnt).
// ---------------------------------------------------------------------------

#define BN   1024                       // batch size B
#define BLK  256                        // threads per block (8 x wave32)
#define JPT  (BN / BLK)                 // output columns per thread = 4
#define SCALE_LOG2 14.426950408889634f  // (1/TAU) * log2(e), TAU = 0.1
#define CLAMP_ARG  115.0f               // keep exp2 result finite in f32
#define KCONST     60.0f

#ifndef __has_builtin
#define __has_builtin(x) 0
#endif

#if __has_builtin(__builtin_amdgcn_global_load_async_to_lds_b128) && \
    __has_builtin(__builtin_amdgcn_global_load_async_to_lds_b32)
#define HAVE_ASYNC_LDS 1
// Exact parameter types per clang diagnostics:
//   b128: (AS1 int __vector_size(16)*, AS3 int __vector_size(16)*, imm, imm)
//   b32 : (AS1 int*,                   AS3 int*,                   imm, imm)
typedef int i32x4_vec __attribute__((vector_size(16)));
typedef __attribute__((address_space(1))) i32x4_vec* gptr_b128;
typedef __attribute__((address_space(3))) i32x4_vec* lptr_b128;
typedef __attribute__((address_space(1))) int*       gptr_b32;
typedef __attribute__((address_space(3))) int*       lptr_b32;
#else
#define HAVE_ASYNC_LDS 0
#endif

#if HAVE_ASYNC_LDS
#if __has_builtin(__builtin_amdgcn_s_wait_asynccnt)
#define WAIT_ASYNCCNT0() __builtin_amdgcn_s_wait_asynccnt(0)
#else
#define WAIT_ASYNCCNT0() asm volatile("s_wait_asynccnt 0" ::: "memory")
#endif
#endif

__device__ __forceinline__ float fast_rcp(float x) {
#if __has_builtin(__builtin_amdgcn_rcpf)
  return __builtin_amdgcn_rcpf(x);   // v_rcp_f32 (TRANS)
#else
  return 1.0f / x;
#endif
}

__device__ __forceinline__ float fast_exp2(float x) {
#if __has_builtin(__builtin_amdgcn_exp2f)
  return __builtin_amdgcn_exp2f(x);  // v_exp_f32 (TRANS)
#else
  return exp2f(x);
#endif
}

__device__ __forceinline__ float exp_row(float s) {
  float t = fminf(fmaxf(s * SCALE_LOG2, -CLAMP_ARG), CLAMP_ARG);
  return fast_exp2(t);
}

__global__ __launch_bounds__(BLK) void
DifferentiableRankIntegration_kernel(const float* __restrict__ s_v,
                                     const float* __restrict__ s_l,
                                     const unsigned char* __restrict__ pos_mask,
                                     const float* __restrict__ w_v,
                                     const float* __restrict__ w_l,
                                     float* __restrict__ out) {
  __shared__ float e_v[BN];                 // exp(s_v[c,:] / tau)
  __shared__ float e_l[BN];                 // exp(s_l[c,:] / tau)
  __shared__ float posf[BN];                // pos_mask[c,:] as 0.0/1.0
  __shared__ unsigned int mstage[BN / 4];   // raw mask bytes staging

  const int t   = (int)threadIdx.x;
  const int row = (int)blockIdx.x * BN;     // this block owns row c = blockIdx.x
  const int k4  = t * 4;                    // this thread's 4 staging slots

  // ---- Phase 1: stage this row into LDS -----------------------------------
  float4 sv4, sl4;
  unsigned int mm;
#if HAVE_ASYNC_LDS
  // gfx1250 async copy: 16B/lane per row + 4 mask bytes/lane (ASYNCcnt).
  __builtin_amdgcn_global_load_async_to_lds_b128(
      (gptr_b128)(s_v + row + k4), (lptr_b128)(&e_v[k4]), 0, 0);
  __builtin_amdgcn_global_load_async_to_lds_b128(
      (gptr_b128)(s_l + row + k4), (lptr_b128)(&e_l[k4]), 0, 0);
  __builtin_amdgcn_global_load_async_to_lds_b32(
      (gptr_b32)(pos_mask + row + k4), (lptr_b32)(&mstage[t]), 0, 0);
  WAIT_ASYNCCNT0();                         // own lanes' copies complete
  sv4 = *reinterpret_cast<const float4*>(&e_v[k4]);
  sl4 = *reinterpret_cast<const float4*>(&e_l[k4]);
  mm  = mstage[t];
#else
  sv4 = *reinterpret_cast<const float4*>(s_v + row + k4);
  sl4 = *reinterpret_cast<const float4*>(s_l + row + k4);
  mm  = *reinterpret_cast<const unsigned int*>(pos_mask + row + k4);
#endif

  // transform in place: e = exp(s/tau), mask byte -> float
  e_v[k4 + 0] = exp_row(sv4.x);  e_v[k4 + 1] = exp_row(sv4.y);
  e_v[k4 + 2] = exp_row(sv4.z);  e_v[k4 + 3] = exp_row(sv4.w);
  e_l[k4 + 0] = exp_row(sl4.x);  e_l[k4 + 1] = exp_row(sl4.y);
  e_l[k4 + 2] = exp_row(sl4.z);  e_l[k4 + 3] = exp_row(sl4.w);
  posf[k4 + 0] = (float)(mm & 0xFFu);
  posf[k4 + 1] = (float)((mm >> 8) & 0xFFu);
  posf[k4 + 2] = (float)((mm >> 16) & 0xFFu);
  posf[k4 + 3] = (float)((mm >> 24) & 0xFFu);
  __syncthreads();

  // ---- Phase 2: pairwise reduction over k ---------------------------------
  float evj[JPT], elj[JPT];
  float aV[JPT] = {}, aVP[JPT] = {}, aL[JPT] = {}, aLP[JPT] = {};
#pragma unroll
  for (int u = 0; u < JPT; ++u) {
    evj[u] = e_v[t + u * BLK];
    elj[u] = e_l[t + u * BLK];
  }

  for (int k = 0; k < BN; k += 4) {
    // LDS broadcast reads (all lanes same address -> conflict free)
    float4 ekv = *reinterpret_cast<const float4*>(&e_v[k]);
    float4 ekl = *reinterpret_cast<const float4*>(&e_l[k]);
    float4 pk  = *reinterpret_cast<const float4*>(&posf[k]);
    const float evk[4] = {ekv.x, ekv.y, ekv.z, ekv.w};
    const float elk[4] = {ekl.x, ekl.y, ekl.z, ekl.w};
    const float pkk[4] = {pk.x, pk.y, pk.z, pk.w};
#pragma unroll
    for (int kk = 0; kk < 4; ++kk) {
#pragma unroll
      for (int u = 0; u < JPT; ++u) {       // 8 independent rcp chains / kk
        float sv = evk[kk] * fast_rcp(evk[kk] + evj[u]);  // sigmoid, matrix v
        float sl = elk[kk] * fast_rcp(elk[kk] + elj[u]);  // sigmoid, matrix l
        aV[u] += sv;  aVP[u] = fmaf(sv, pkk[kk], aVP[u]);
        aL[u] += sl;  aLP[u] = fmaf(sl, pkk[kk], aLP[u]);
      }
    }
  }

  // ---- Phase 3: ranks + fused S_hat combine -------------------------------
#pragma unroll
  for (int u = 0; u < JPT; ++u) {
    int   j  = t + u * BLK;
    float pj = posf[j];
    // rank_pos = 1 + (sumSig - sumSigPos);  rank_neg = 1 + sumSigPos
    float rank_v = pj * (1.0f + aV[u] - aVP[u]) + (1.0f - pj) * (1.0f + aVP[u]);
    float rank_l = pj * (1.0f + aL[u] - aLP[u]) + (1.0f - pj) * (1.0f + aLP[u]);
    float o = (KCONST + 1.0f) *
              (w_v[row + j] * fast_rcp(KCONST + rank_v) +
               w_l[row + j] * fast_rcp(KCONST + rank_l));
    out[row + j] = o;
  }
}

extern "C" void kernel_launch(void* const* d_in, const int* in_sizes, int n_in,
                              void* d_out, int out_size, void* d_ws, size_t ws_size,
                              hipStream_t stream) {
  (void)in_sizes; (void)n_in; (void)out_size; (void)d_ws; (void)ws_size;
  // setup_inputs order: s_v, s_l, pos_mask, neg_mask, w_v, w_l
  const float*         s_v = (const float*)d_in[0];
  const float*         s_l = (const float*)d_in[1];
  const unsigned char* pos = (const unsigned char*)d_in[2];  // jax bool = 1 byte
  // d_in[3] (neg_mask) is the exact complement of pos_mask -> not needed
  const float*         w_v = (const float*)d_in[4];
  const float*         w_l = (const float*)d_in[5];
  float*               out = (float*)d_out;

  dim3 grid(BN), block(BLK);  // one block per row, 8 wave32 per block
  hipLaunchKernelGGL(DifferentiableRankIntegration_kernel, grid, block, 0, stream,
                     s_v, s_l, pos, w_v, w_l, out);
}